// SubstructureAttention_57011395887500
// MI455X (gfx1250) — compile-verified
//
#include <hip/hip_runtime.h>
#include <hip/hip_bf16.h>

// ---------------------------------------------------------------------------
// SubstructureAttention for MI455X (gfx1250, wave32)
//   pass1: streaming segment-sum; explicit global_atomic_add_f32 (DEV scope,
//          non-returning, resolves in the 192MB GL2 -> table is L2-resident)
//   pass2: means -> tanh(means@W1+b1)@w2 via V_WMMA_F32_16X16X4_F32
//   pass3: softmax over 8192 scores (1 block)
//   pass4: gather attention back to atoms (vectorized)
// ---------------------------------------------------------------------------

typedef __attribute__((ext_vector_type(2))) float v2f;
typedef __attribute__((ext_vector_type(8))) float v8f;

#define FEAT 128
#define HID  64
#define SEGS 8192

__device__ __forceinline__ void atomic_add_f32(float* p, float v) {
    // Non-returning FP32 global atomic, device scope: RMW executes in GL2.
    // Tracked with STOREcnt; kernel-end implicit wait-idle guarantees
    // completion before the dispatch completes.
    asm volatile("global_atomic_add_f32 %0, %1, off scope:SCOPE_DEV"
                 :: "v"(p), "v"(v)
                 : "memory");
}

// ---- zero scratch -----------------------------------------------------------
__global__ void sa_zero_kernel(float* __restrict__ p, int n) {
    int i = blockIdx.x * blockDim.x + threadIdx.x;
    if (i < n) p[i] = 0.0f;
}

// ---- pass 1: segment sums + counts -----------------------------------------
// One wave (32 lanes) per atom row; each lane handles 4 consecutive features
// (float4 load -> 512B contiguous per wave). subst_id is wave-uniform.
__global__ __launch_bounds__(256) void sa_segsum_kernel(
        const float* __restrict__ x, const int* __restrict__ ids,
        float* __restrict__ seg_sum, float* __restrict__ counts, int natoms) {
    int t    = blockIdx.x * blockDim.x + threadIdx.x; // up to 64M, fits int
    int atom = t >> 5;
    int q    = t & 31;                                // feature quad 0..31
    if (atom >= natoms) return;
    int sid = ids[atom];
    const float4 v = ((const float4*)(x + (size_t)atom * FEAT))[q];
    float* dst = seg_sum + (size_t)sid * FEAT + q * 4;
    atomic_add_f32(dst + 0, v.x);
    atomic_add_f32(dst + 1, v.y);
    atomic_add_f32(dst + 2, v.z);
    atomic_add_f32(dst + 3, v.w);
    if (q == 0) atomic_add_f32(&counts[sid], 1.0f);
}

// ---- pass 2: means -> MLP scores via WMMA f32 16x16x4 -----------------------
// One wave per 16-segment tile. C tile = 16 rows x 16 cols; N=64 -> 4 col
// tiles; K=128 -> 32 chunks of 4.
// A layout (16x4, MxK):  lane<16: M=lane, vgpr0=K0, vgpr1=K1
//                        lane>=16: M=lane-16, vgpr0=K2, vgpr1=K3
// B layout (4x16, KxN):  lane<16: N=lane, vgpr0=K0, vgpr1=K1
//                        lane>=16: N=lane-16, vgpr0=K2, vgpr1=K3
// C layout: vgpr r -> M=r (lanes 0-15) / M=8+r (lanes 16-31); N=lane&15
__global__ __launch_bounds__(32) void sa_mlp_kernel(
        const float* __restrict__ seg_sum, const float* __restrict__ counts,
        const float* __restrict__ W1, const float* __restrict__ b1,
        const float* __restrict__ w2, float* __restrict__ scores) {
    const int tile = blockIdx.x;         // 0..511
    const int lane = threadIdx.x;        // 0..31
    const int half = lane >> 4;          // 0|1
    const int l16  = lane & 15;
    const int row  = tile * 16 + l16;    // this lane's A-matrix row (M = l16)

    const float inv = 1.0f / fmaxf(counts[row], 1.0f);

    float sacc[8];
#pragma unroll
    for (int r = 0; r < 8; ++r) sacc[r] = 0.0f;

    for (int nt = 0; nt < 4; ++nt) {
        v8f c = {};
#pragma unroll 8
        for (int kc = 0; kc < 32; ++kc) {
            const int k0 = kc * 4 + half * 2;
            v2f a, b;
            a.x = seg_sum[(size_t)row * FEAT + k0 + 0] * inv;  // mean on the fly
            a.y = seg_sum[(size_t)row * FEAT + k0 + 1] * inv;
            const int colB = nt * 16 + l16;
            b.x = W1[(k0 + 0) * HID + colB];
            b.y = W1[(k0 + 1) * HID + colB];
            c = __builtin_amdgcn_wmma_f32_16x16x4_f32(
                    /*neg_a=*/false, a, /*neg_b=*/false, b,
                    /*c_mod=*/(short)0, c, /*reuse_a=*/false, /*reuse_b=*/false);
        }
        // bias + tanh + partial dot with w2 (this lane's column)
        const int col  = nt * 16 + l16;
        const float bb = b1[col];
        const float wc = w2[col];
#pragma unroll
        for (int r = 0; r < 8; ++r) {
            sacc[r] += tanhf(c[r] + bb) * wc;
        }
    }
    // reduce over the 16 lanes of each half (sums over N); xor masks 1..8
    // never cross the half boundary (bit 4).
#pragma unroll
    for (int off = 8; off >= 1; off >>= 1) {
#pragma unroll
        for (int r = 0; r < 8; ++r)
            sacc[r] += __shfl_xor(sacc[r], off, 32);
    }
    if (l16 == 0) {
#pragma unroll
        for (int r = 0; r < 8; ++r)
            scores[tile * 16 + half * 8 + r] = sacc[r];  // m = r + 8*half
    }
}

// ---- pass 3: softmax over SEGS scores (single block) ------------------------
__global__ __launch_bounds__(1024) void sa_softmax_kernel(
        const float* __restrict__ scores, float* __restrict__ attn) {
    __shared__ float sm[1024];
    const int t = threadIdx.x;

    float m = -3.402823466e38f;
    for (int i = t; i < SEGS; i += 1024) m = fmaxf(m, scores[i]);
    sm[t] = m;
    __syncthreads();
    for (int s = 512; s > 0; s >>= 1) {
        if (t < s) sm[t] = fmaxf(sm[t], sm[t + s]);
        __syncthreads();
    }
    const float gmax = sm[0];
    __syncthreads();

    float sum = 0.0f;
    for (int i = t; i < SEGS; i += 1024) sum += expf(scores[i] - gmax);
    sm[t] = sum;
    __syncthreads();
    for (int s = 512; s > 0; s >>= 1) {
        if (t < s) sm[t] += sm[t + s];
        __syncthreads();
    }
    const float inv = 1.0f / sm[0];

    for (int i = t; i < SEGS; i += 1024)
        attn[i] = expf(scores[i] - gmax) * inv;
}

// ---- pass 4: gather per-atom attention -------------------------------------
__global__ __launch_bounds__(256) void sa_gather_kernel(
        const int* __restrict__ ids, const float* __restrict__ attn,
        float* __restrict__ out, int natoms) {
    const int i = blockIdx.x * blockDim.x + threadIdx.x; // one quad per thread
    const int b = i * 4;
    if (b + 3 < natoms) {
        const int4 id = ((const int4*)ids)[i];
        float4 o;
        o.x = attn[id.x];
        o.y = attn[id.y];
        o.z = attn[id.z];
        o.w = attn[id.w];
        ((float4*)out)[i] = o;
    } else if (b < natoms) {
        for (int j = b; j < natoms; ++j) out[j] = attn[ids[j]];
    }
}

// ---------------------------------------------------------------------------
extern "C" void kernel_launch(void* const* d_in, const int* in_sizes, int n_in,
                              void* d_out, int out_size, void* d_ws, size_t ws_size,
                              hipStream_t stream) {
    const float* x   = (const float*)d_in[0];   // [N,128]
    const int*   ids = (const int*)d_in[1];     // [N]
    // d_in[2] = num_segments scalar (fixed 8192 in this problem)
    const float* W1  = (const float*)d_in[3];   // [128,64]
    const float* b1  = (const float*)d_in[4];   // [64]
    const float* w2  = (const float*)d_in[5];   // [64,1]
    float* out = (float*)d_out;                 // [N,1]

    const int natoms = in_sizes[0] / FEAT;      // 2,000,000

    // workspace layout (~4.3 MB)
    float* seg_sum = (float*)d_ws;              // SEGS*FEAT
    float* counts  = seg_sum + (size_t)SEGS * FEAT;
    float* scores  = counts + SEGS;
    float* attn    = scores + SEGS;

    // 1) zero accumulators
    const int nz = SEGS * FEAT + SEGS;
    sa_zero_kernel<<<(nz + 255) / 256, 256, 0, stream>>>(seg_sum, nz);

    // 2) segment sums + counts (one wave per atom row)
    const long long tot = (long long)natoms * 32;
    sa_segsum_kernel<<<(int)((tot + 255) / 256), 256, 0, stream>>>(
        x, ids, seg_sum, counts, natoms);

    // 3) means -> MLP scores via WMMA (one wave per 16 segments)
    sa_mlp_kernel<<<SEGS / 16, 32, 0, stream>>>(seg_sum, counts, W1, b1, w2, scores);

    // 4) softmax over segments
    sa_softmax_kernel<<<1, 1024, 0, stream>>>(scores, attn);

    // 5) gather to atoms
    const int nquads = (natoms + 3) / 4;
    sa_gather_kernel<<<(nquads + 255) / 256, 256, 0, stream>>>(ids, attn, out, natoms);
}